// ChebEncoder_82781199663546
// MI455X (gfx1250) — compile-verified
//
#include <hip/hip_runtime.h>
#include <hip/hip_bf16.h>

typedef __attribute__((ext_vector_type(2))) float v2f;
typedef __attribute__((ext_vector_type(8))) float v8f;

#define EPS 1e-5f
#define SLOPE 0.01f

// ---------------------------------------------------------------------------
// Graph preprocessing
// ---------------------------------------------------------------------------
__global__ void degree_k(const int* __restrict__ row, float* __restrict__ deg, int nE) {
    int e = blockIdx.x * blockDim.x + threadIdx.x;
    if (e < nE) unsafeAtomicAdd(&deg[row[e]], 1.0f);
}

__global__ void dinv_k(float* __restrict__ deg, int n) {
    int i = blockIdx.x * blockDim.x + threadIdx.x;
    if (i < n) {
        float d = deg[i];
        deg[i] = (d > 0.0f) ? rsqrtf(d) : 0.0f;
    }
}

__global__ void edgew_k(const int* __restrict__ row, const int* __restrict__ col,
                        const float* __restrict__ dinv, float* __restrict__ ew, int nE) {
    int e = blockIdx.x * blockDim.x + threadIdx.x;
    if (e < nE) ew[e] = -dinv[row[e]] * dinv[col[e]];
}

// ---------------------------------------------------------------------------
// SpMM: out[col[e], :] += ew[e] * h[row[e], :]  (one wave per edge, lanes = channels)
// ---------------------------------------------------------------------------
__global__ void spmm_scatter(const float* __restrict__ h, const float* __restrict__ ew,
                             const int* __restrict__ row, const int* __restrict__ col,
                             float* __restrict__ out, int nE, int C) {
    int wid = (blockIdx.x * blockDim.x + threadIdx.x) >> 5;
    int lane = threadIdx.x & 31;
    if (wid >= nE) return;
    int s = row[wid];
    int d = col[wid];
    float w = ew[wid];
    const float* hs = h + (size_t)s * C;
    float* od = out + (size_t)d * C;
    for (int c = lane; c < C; c += 32) {
        unsafeAtomicAdd(&od[c], w * hs[c]);
    }
}

// ---------------------------------------------------------------------------
// Chebyshev GEMM via fp32 WMMA with LDS-staged weights and fused BN stats:
//   Out[N,Cout] = A0[N,Cin] @ W[0] + A1[N,Cin] @ W[1] + bias
// Each wave owns a 16-row strip and computes all NT = Cout/16 column tiles,
// reusing each A fragment across NT wmmas. W (stacked 2*Cin x Cout, <=64KB)
// is staged into LDS once per block with async global->LDS loads.
// ---------------------------------------------------------------------------
template <int NT>
__global__ void gemm_cheb(const float* __restrict__ A0, const float* __restrict__ A1,
                          const float* __restrict__ W, const float* __restrict__ bias,
                          float* __restrict__ Out, float* __restrict__ sums,
                          float* __restrict__ sqs, int nRows, int Cin, int Cout,
                          int doStats) {
    extern __shared__ float sW[];   // [2*Cin, Cout]
    const int tid = threadIdx.x;
    const int nW = 2 * Cin * Cout;

    // ---- async-copy stacked weights into LDS (ASYNCcnt-tracked) ------------
    {
        unsigned ldsBase = __builtin_amdgcn_groupstaticsize();  // dynamic LDS origin
        for (int i = tid * 4; i < nW; i += blockDim.x * 4) {
            unsigned lds_off = ldsBase + (unsigned)i * 4u;
            unsigned long long ga = (unsigned long long)(const void*)(W + i);
            asm volatile("global_load_async_to_lds_b128 %0, %1, off"
                         :: "v"(lds_off), "v"(ga) : "memory");
        }
        asm volatile("s_wait_asynccnt 0x0" ::: "memory");
    }
    __syncthreads();

    const int lane = tid & 31;
    const int wave = tid >> 5;
    const int mT = blockIdx.x * (blockDim.x >> 5) + wave;  // one 16-row strip per wave
    if (mT >= (nRows >> 4)) return;   // wave-uniform: EXEC stays all-ones
    const int m0 = mT << 4;
    const int half = lane >> 4;       // 0: K pair {0,1}, 1: K pair {2,3}
    const int lr   = lane & 15;

    const float* arow0 = A0 + (size_t)(m0 + lr) * Cin;   // A frag: M = lr
    const float* arow1 = A1 + (size_t)(m0 + lr) * Cin;
    const float* sWa = sW;                               // Cheb order-0 rows
    const float* sWb = sW + (size_t)Cin * Cout;          // Cheb order-1 rows

    v8f acc[NT] = {};
    // ---- K loop over Tx0 = h ------------------------------------------------
    for (int kb = 0; kb < Cin; kb += 4) {
        v2f a;
        const float* ap = arow0 + kb + 2 * half;
        a.x = ap[0];
        a.y = ap[1];
        const float* wr = sWa + (size_t)(kb + 2 * half) * Cout + lr;
#pragma unroll
        for (int nt = 0; nt < NT; ++nt) {
            v2f b;
            b.x = wr[nt * 16];
            b.y = wr[Cout + nt * 16];
            acc[nt] = __builtin_amdgcn_wmma_f32_16x16x4_f32(false, a, false, b,
                                                            (short)0, acc[nt],
                                                            false, false);
        }
    }
    // ---- K loop over Tx1 = L_hat @ h ----------------------------------------
    for (int kb = 0; kb < Cin; kb += 4) {
        v2f a;
        const float* ap = arow1 + kb + 2 * half;
        a.x = ap[0];
        a.y = ap[1];
        const float* wr = sWb + (size_t)(kb + 2 * half) * Cout + lr;
#pragma unroll
        for (int nt = 0; nt < NT; ++nt) {
            v2f b;
            b.x = wr[nt * 16];
            b.y = wr[Cout + nt * 16];
            acc[nt] = __builtin_amdgcn_wmma_f32_16x16x4_f32(false, a, false, b,
                                                            (short)0, acc[nt],
                                                            false, false);
        }
    }
    // ---- epilogue: bias add, store, fused BN partial stats ------------------
#pragma unroll
    for (int nt = 0; nt < NT; ++nt) {
        const int ncol = nt * 16 + lr;
        const float bv = bias[ncol];
        float s = 0.0f, sq = 0.0f;
#pragma unroll
        for (int r = 0; r < 8; ++r) {
            // C/D layout: VGPR r, lanes 0-15 -> M = r ; lanes 16-31 -> M = r+8
            float v = acc[nt][r] + bv;
            Out[(size_t)(m0 + r + 8 * half) * Cout + ncol] = v;
            s += v;
            sq += v * v;
        }
        if (doStats) {
            s  += __shfl_xor(s, 16, 32);    // combine the two lane halves
            sq += __shfl_xor(sq, 16, 32);
            if (half == 0) {
                unsafeAtomicAdd(&sums[ncol], s);
                unsafeAtomicAdd(&sqs[ncol], sq);
            }
        }
    }
}

// ---------------------------------------------------------------------------
// BatchNorm finalize + apply
// ---------------------------------------------------------------------------
__global__ void bn_finalize(const float* __restrict__ sums, const float* __restrict__ sqs,
                            const float* __restrict__ gamma, const float* __restrict__ beta,
                            float* __restrict__ ca, float* __restrict__ cb, int n, int C) {
    int c = blockIdx.x * blockDim.x + threadIdx.x;
    if (c < C) {
        float invn = 1.0f / (float)n;
        float m = sums[c] * invn;
        float v = sqs[c] * invn - m * m;
        float r = rsqrtf(v + EPS);
        float sa = gamma[c] * r;
        ca[c] = sa;
        cb[c] = beta[c] - m * sa;
    }
}

__global__ void bn_apply_lrelu(float* __restrict__ h, const float* __restrict__ ca,
                               const float* __restrict__ cb, int total, int C) {
    int i = blockIdx.x * blockDim.x + threadIdx.x;
    if (i < total) {
        int c = i & (C - 1);
        float v = h[i] * ca[c] + cb[c];
        h[i] = (v >= 0.0f) ? v : SLOPE * v;
    }
}

// ---------------------------------------------------------------------------
// Orchestration
// ---------------------------------------------------------------------------
extern "C" void kernel_launch(void* const* d_in, const int* in_sizes, int n_in,
                              void* d_out, int out_size, void* d_ws, size_t ws_size,
                              hipStream_t stream) {
    const float* x   = (const float*)d_in[0];
    const int*   ei  = (const int*)d_in[1];
    const float* W1  = (const float*)d_in[2];
    const float* b1  = (const float*)d_in[3];
    const float* W2  = (const float*)d_in[4];
    const float* b2  = (const float*)d_in[5];
    const float* W3  = (const float*)d_in[6];
    const float* b3  = (const float*)d_in[7];
    const float* g1  = (const float*)d_in[8];
    const float* be1 = (const float*)d_in[9];
    const float* g2  = (const float*)d_in[10];
    const float* be2 = (const float*)d_in[11];

    const int IN_C = 64, HID = 128, HID2 = 64;
    const int N = in_sizes[0] / IN_C;
    const int E = in_sizes[1] / 2;
    const int* row = ei;
    const int* col = ei + E;

    // ---- workspace carve-out -------------------------------------------------
    char* ws = (char*)d_ws;
    size_t off = 0;
    auto carve = [&](size_t bytes) -> void* {
        void* p = ws + off;
        off = (off + bytes + 255) & ~(size_t)255;
        return p;
    };
    float* dinv = (float*)carve((size_t)N * 4);            // degree -> rsqrt in place
    float* ew   = (float*)carve((size_t)E * 4);            // edge weights of L_hat
    float* t    = (float*)carve((size_t)N * HID * 4);      // prop() scratch (max C=128)
    float* h1   = (float*)carve((size_t)N * HID * 4);      // layer-1 activations
    float* h2   = (float*)carve((size_t)N * HID2 * 4);     // layer-2 activations
    float* sums = (float*)carve(HID * 4);
    float* sqs  = (float*)carve(HID * 4);
    float* ca   = (float*)carve(HID * 4);
    float* cb   = (float*)carve(HID * 4);
    (void)ws_size; (void)n_in; (void)out_size;

    const int TPB = 256;
    const int eBlocks  = (E + TPB - 1) / TPB;
    const int nBlocks  = (N + TPB - 1) / TPB;
    const int spBlocks = (E + 7) / 8;               // 8 waves/block, 1 edge/wave
    const int gBlocks  = ((N >> 4) + 7) / 8;        // 8 M-strips per block

    // ---- graph preprocessing -------------------------------------------------
    hipMemsetAsync(dinv, 0, (size_t)N * 4, stream);
    degree_k<<<eBlocks, TPB, 0, stream>>>(row, dinv, E);
    dinv_k<<<nBlocks, TPB, 0, stream>>>(dinv, N);
    edgew_k<<<eBlocks, TPB, 0, stream>>>(row, col, dinv, ew, E);

    // ---- layer 1: 64 -> 128, BN, LeakyReLU ----------------------------------
    hipMemsetAsync(t, 0, (size_t)N * IN_C * 4, stream);
    spmm_scatter<<<spBlocks, TPB, 0, stream>>>(x, ew, row, col, t, E, IN_C);
    hipMemsetAsync(sums, 0, HID * 4, stream);
    hipMemsetAsync(sqs, 0, HID * 4, stream);
    gemm_cheb<8><<<gBlocks, TPB, 2 * IN_C * HID * 4, stream>>>(
        x, t, W1, b1, h1, sums, sqs, N, IN_C, HID, 1);
    bn_finalize<<<1, HID, 0, stream>>>(sums, sqs, g1, be1, ca, cb, N, HID);
    bn_apply_lrelu<<<((size_t)N * HID + TPB - 1) / TPB, TPB, 0, stream>>>(
        h1, ca, cb, N * HID, HID);

    // ---- layer 2: 128 -> 64, BN, LeakyReLU ----------------------------------
    hipMemsetAsync(t, 0, (size_t)N * HID * 4, stream);
    spmm_scatter<<<spBlocks, TPB, 0, stream>>>(h1, ew, row, col, t, E, HID);
    hipMemsetAsync(sums, 0, HID2 * 4, stream);
    hipMemsetAsync(sqs, 0, HID2 * 4, stream);
    gemm_cheb<4><<<gBlocks, TPB, 2 * HID * HID2 * 4, stream>>>(
        h1, t, W2, b2, h2, sums, sqs, N, HID, HID2, 1);
    bn_finalize<<<1, HID2, 0, stream>>>(sums, sqs, g2, be2, ca, cb, N, HID2);
    bn_apply_lrelu<<<((size_t)N * HID2 + TPB - 1) / TPB, TPB, 0, stream>>>(
        h2, ca, cb, N * HID2, HID2);

    // ---- layer 3: 64 -> 64 (no BN / activation) -----------------------------
    hipMemsetAsync(t, 0, (size_t)N * HID2 * 4, stream);
    spmm_scatter<<<spBlocks, TPB, 0, stream>>>(h2, ew, row, col, t, E, HID2);
    gemm_cheb<4><<<gBlocks, TPB, 2 * HID2 * IN_C * 4, stream>>>(
        h2, t, W3, b3, (float*)d_out, sums, sqs, N, HID2, IN_C, 0);
}